// OTLoss_81295140979398
// MI455X (gfx1250) — compile-verified
//
#include <hip/hip_runtime.h>
#include <math.h>

typedef __attribute__((ext_vector_type(2))) float v2f;
typedef __attribute__((ext_vector_type(8))) float v8f;

#define NROWS   4096
#define FEAT    128
#define KT      1024
#define KS      2048
#define NTOT    4096
#define GAMMA_  1.0f
#define EPS_    1e-6f
#define MAXIT_  20

__device__ __forceinline__ v8f wmma_f32_16x16x4(v2f a, v2f b, v8f c) {
  // 8 args: (neg_a, A, neg_b, B, c_mod, C, reuse_a, reuse_b)
  return __builtin_amdgcn_wmma_f32_16x16x4_f32(false, a, false, b, (short)0, c,
                                               false, false);
}

// ---------------------------------------------------------------------------
// E[r,f] = sum_k X[r,k]*W[f,k] + b[f]  (X: NROWS x K, W: FEAT x K row-major)
// 2x2 register-blocked: each wave owns a 32x32 output block (4 accumulators).
// Per K-step: 4 loads -> 4 WMMAs, with A/B fragment reuse.
// ---------------------------------------------------------------------------
template <int K>
__global__ void embed_gemm_kernel(const float* __restrict__ X,
                                  const float* __restrict__ W,
                                  const float* __restrict__ bias,
                                  float* __restrict__ E) {
  const int lane = threadIdx.x & 31;
  const int wid  = blockIdx.x * (blockDim.x >> 5) + (threadIdx.x >> 5);
  const int nb   = FEAT / 32;                    // 4 N-blocks
  const int bm   = wid / nb;
  const int bn   = wid % nb;
  const int khalf = (lane >> 4) * 2;             // K sub-pair: 0 or 2

  const float* A0 = X + (size_t)(bm * 32 + (lane & 15)) * K;
  const float* A1 = A0 + (size_t)16 * K;
  const float* B0 = W + (size_t)(bn * 32 + (lane & 15)) * K;
  const float* B1 = B0 + (size_t)16 * K;

  v8f acc00 = {}, acc01 = {}, acc10 = {}, acc11 = {};
#pragma unroll 4
  for (int k = 0; k < K; k += 4) {
    v2f a0 = *(const v2f*)(A0 + k + khalf);
    v2f a1 = *(const v2f*)(A1 + k + khalf);
    v2f b0 = *(const v2f*)(B0 + k + khalf);
    v2f b1 = *(const v2f*)(B1 + k + khalf);
    acc00 = wmma_f32_16x16x4(a0, b0, acc00);
    acc01 = wmma_f32_16x16x4(a0, b1, acc01);
    acc10 = wmma_f32_16x16x4(a1, b0, acc10);
    acc11 = wmma_f32_16x16x4(a1, b1, acc11);
  }
  const int   c0  = bn * 32 + (lane & 15), c1 = c0 + 16;
  const float bv0 = bias[c0], bv1 = bias[c1];
  const int   r0  = bm * 32 + ((lane >> 4) << 3);   // +8 for upper half-wave
  const int   r1  = r0 + 16;
#pragma unroll
  for (int v = 0; v < 8; ++v) {
    E[(size_t)(r0 + v) * FEAT + c0] = acc00[v] + bv0;
    E[(size_t)(r0 + v) * FEAT + c1] = acc01[v] + bv1;
    E[(size_t)(r1 + v) * FEAT + c0] = acc10[v] + bv0;
    E[(size_t)(r1 + v) * FEAT + c1] = acc11[v] + bv1;
  }
}

// Row-normalize E (4096 x 128): one block of 128 threads per row.
__global__ void rownorm_kernel(float* __restrict__ E) {
  __shared__ float red[128];
  const int r = blockIdx.x, t = threadIdx.x;
  float v = E[(size_t)r * FEAT + t];
  red[t] = v * v;
  __syncthreads();
  for (int s = 64; s > 0; s >>= 1) {
    if (t < s) red[t] += red[t + s];
    __syncthreads();
  }
  const float inv = 1.0f / sqrtf(red[0]);
  E[(size_t)r * FEAT + t] = v * inv;
}

// ---------------------------------------------------------------------------
// Mout[i,j] = 1 - dot(ET[i,:], ES[j,:])  (4096 x 4096, K = 128).
// 2x4 register-blocked: each wave owns a 32x64 output block (8 accumulators).
// Per K-step: 6 loads -> 8 WMMAs.
// ---------------------------------------------------------------------------
__global__ void sim_gemm_kernel(const float* __restrict__ ET,
                                const float* __restrict__ ES,
                                float* __restrict__ Mout) {
  const int lane = threadIdx.x & 31;
  const int wid  = blockIdx.x * (blockDim.x >> 5) + (threadIdx.x >> 5);
  const int nb   = NTOT / 64;                    // 64 N-blocks
  const int bm   = wid / nb;
  const int bn   = wid % nb;
  const int khalf = (lane >> 4) * 2;

  const float* A0 = ET + (size_t)(bm * 32 + (lane & 15)) * FEAT;
  const float* A1 = A0 + (size_t)16 * FEAT;
  const float* Bp[4];
#pragma unroll
  for (int n = 0; n < 4; ++n)
    Bp[n] = ES + (size_t)(bn * 64 + n * 16 + (lane & 15)) * FEAT;

  v8f acc[2][4];
#pragma unroll
  for (int m = 0; m < 2; ++m)
#pragma unroll
    for (int n = 0; n < 4; ++n) acc[m][n] = (v8f){};

#pragma unroll 4
  for (int k = 0; k < FEAT; k += 4) {
    v2f a0 = *(const v2f*)(A0 + k + khalf);
    v2f a1 = *(const v2f*)(A1 + k + khalf);
    v2f b[4];
#pragma unroll
    for (int n = 0; n < 4; ++n) b[n] = *(const v2f*)(Bp[n] + k + khalf);
#pragma unroll
    for (int n = 0; n < 4; ++n) {
      acc[0][n] = wmma_f32_16x16x4(a0, b[n], acc[0][n]);
      acc[1][n] = wmma_f32_16x16x4(a1, b[n], acc[1][n]);
    }
  }
  const int r0 = bm * 32 + ((lane >> 4) << 3);
#pragma unroll
  for (int m = 0; m < 2; ++m) {
    const int rbase = r0 + m * 16;
#pragma unroll
    for (int n = 0; n < 4; ++n) {
      const int col = bn * 64 + n * 16 + (lane & 15);
#pragma unroll
      for (int v = 0; v < 8; ++v)
        Mout[(size_t)(rbase + v) * NTOT + col] = 1.0f - acc[m][n][v];
    }
  }
}

// ---------------------------------------------------------------------------
// P0 = exp(-GAMMA*(M - rowmin(M)));  also alpha[i] = rowsum(P0).
// ---------------------------------------------------------------------------
__global__ void p0_kernel(const float* __restrict__ Mmat,
                          float* __restrict__ P, float* __restrict__ alpha) {
  __shared__ float red[256];
  const int i = blockIdx.x, t = threadIdx.x;
  const float* mrow = Mmat + (size_t)i * NTOT;
  float mn = 3.4e38f;
  for (int j = t; j < NTOT; j += 256) mn = fminf(mn, mrow[j]);
  red[t] = mn; __syncthreads();
  for (int s = 128; s > 0; s >>= 1) {
    if (t < s) red[t] = fminf(red[t], red[t + s]);
    __syncthreads();
  }
  const float rowmin = red[0];
  __syncthreads();
  float sum = 0.0f;
  float* prow = P + (size_t)i * NTOT;
  for (int j = t; j < NTOT; j += 256) {
    float p = expf(-GAMMA_ * (mrow[j] - rowmin));
    prow[j] = p;
    sum += p;
  }
  red[t] = sum; __syncthreads();
  for (int s = 128; s > 0; s >>= 1) {
    if (t < s) red[t] += red[t + s];
    __syncthreads();
  }
  if (t == 0) alpha[i] = red[0];
}

// Partial column sums: bpart[rb][j] = sum_{i in rowblock rb} P[i,j]/(alpha[i] or 1)
__global__ void colsum_kernel(const float* __restrict__ P,
                              const float* __restrict__ alpha,
                              float* __restrict__ bpart, int useAlpha) {
  const int j  = blockIdx.x * blockDim.x + threadIdx.x;
  const int rb = blockIdx.y;
  const int i0 = rb * (NTOT / 64);
  float s = 0.0f;
  for (int i = i0; i < i0 + (NTOT / 64); ++i) {
    float p = P[(size_t)i * NTOT + j];
    s += useAlpha ? (p / alpha[i]) : p;
  }
  bpart[(size_t)rb * NTOT + j] = s;
}

// beta[j] = scale * sum_rb bpart[rb][j];  blockmax[blk] = max_j |beta[j]-c|
__global__ void beta_reduce_kernel(const float* __restrict__ bpart,
                                   float* __restrict__ beta,
                                   float* __restrict__ blockmax, float scale) {
  __shared__ float red[256];
  const int t = threadIdx.x;
  const int j = blockIdx.x * 256 + t;
  float s = 0.0f;
  for (int rb = 0; rb < 64; ++rb) s += bpart[(size_t)rb * NTOT + j];
  const float b = s * scale;
  beta[j] = b;
  red[t] = fabsf(b - 1.0f / (float)NTOT);
  __syncthreads();
  for (int st = 128; st > 0; st >>= 1) {
    if (t < st) red[t] = fmaxf(red[t], red[t + st]);
    __syncthreads();
  }
  if (t == 0) blockmax[blockIdx.x] = red[0];
}

__global__ void init_flags_kernel(int* __restrict__ flags) {
  if (threadIdx.x == 0) { flags[0] = 0; flags[1] = 0; flags[2] = 0; }
}

// flags[0]=done carry, flags[1]=done_prev (this iter), flags[2]=newly_done
__global__ void flag_kernel(const float* __restrict__ blockmax,
                            int* __restrict__ flags) {
  if (threadIdx.x == 0) {
    float m = 0.0f;
    for (int b = 0; b < 16; ++b) m = fmaxf(m, blockmax[b]);
    const int newly = (m <= EPS_) ? 1 : 0;
    const int done_prev = flags[0];
    flags[1] = done_prev;
    flags[2] = newly;
    flags[0] = done_prev | newly;
  }
}

// P_next = done_prev ? P : (newly ? P1 : P1*c/beta[j]),  P1 = P*r/alpha[i]
// Fused with next iteration's row-sum: alpha[i] = rowsum(P_next).
__global__ void update_kernel(float* __restrict__ P, float* __restrict__ alpha,
                              const float* __restrict__ beta,
                              const int* __restrict__ flags) {
  __shared__ float red[256];
  const int i = blockIdx.x, t = threadIdx.x;
  const int done_prev = flags[1];
  const int newly     = flags[2];
  const float a  = alpha[i];
  const float r  = 1.0f / (float)NTOT;
  const float c  = 1.0f / (float)NTOT;
  const float s1 = r / a;
  float* prow = P + (size_t)i * NTOT;
  float sum = 0.0f;
  for (int j = t; j < NTOT; j += 256) {
    const float p  = prow[j];
    const float p1 = p * s1;
    float pn;
    if (done_prev)      pn = p;
    else if (newly)     pn = p1;
    else                pn = p1 * (c / beta[j]);
    prow[j] = pn;
    sum += pn;
  }
  red[t] = sum; __syncthreads();
  for (int s = 128; s > 0; s >>= 1) {
    if (t < s) red[t] += red[t + s];
    __syncthreads();
  }
  if (t == 0) alpha[i] = red[0];
}

// Pout = P / colsum;  lossp[i] = sum_j (Pout[i,j] - I[i,j])^2
__global__ void finalize_kernel(const float* __restrict__ P,
                                const float* __restrict__ colsum,
                                float* __restrict__ Pout,
                                float* __restrict__ lossp) {
  __shared__ float red[256];
  const int i = blockIdx.x, t = threadIdx.x;
  const float* prow = P + (size_t)i * NTOT;
  float sum = 0.0f;
  for (int j = t; j < NTOT; j += 256) {
    const float pn = prow[j] / colsum[j];
    Pout[(size_t)i * NTOT + j] = pn;
    const float d = pn - ((j == i) ? 1.0f : 0.0f);
    sum += d * d;
  }
  red[t] = sum; __syncthreads();
  for (int s = 128; s > 0; s >>= 1) {
    if (t < s) red[t] += red[t + s];
    __syncthreads();
  }
  if (t == 0) lossp[i] = red[0];
}

__global__ void loss_kernel(const float* __restrict__ lossp,
                            float* __restrict__ out) {
  __shared__ float red[256];
  const int t = threadIdx.x;
  float s = 0.0f;
  for (int j = t; j < NTOT; j += 256) s += lossp[j];
  red[t] = s; __syncthreads();
  for (int st = 128; st > 0; st >>= 1) {
    if (t < st) red[t] += red[t + st];
    __syncthreads();
  }
  if (t == 0) out[0] = sqrtf(red[0]);
}

// ---------------------------------------------------------------------------
extern "C" void kernel_launch(void* const* d_in, const int* in_sizes, int n_in,
                              void* d_out, int out_size, void* d_ws,
                              size_t ws_size, hipStream_t stream) {
  (void)in_sizes; (void)n_in; (void)out_size; (void)ws_size;
  const float* f_t = (const float*)d_in[0];
  const float* f_s = (const float*)d_in[1];
  const float* Wt  = (const float*)d_in[2];
  const float* bt  = (const float*)d_in[3];
  const float* Ws  = (const float*)d_in[4];
  const float* bs  = (const float*)d_in[5];

  float* out  = (float*)d_out;
  float* loss = out;
  float* Pout = out + 1;
  float* Mout = out + 1 + (size_t)NTOT * NTOT;

  float* ws       = (float*)d_ws;
  float* et       = ws;                                   // 4096*128
  float* es       = et + (size_t)NROWS * FEAT;            // 4096*128
  float* P        = es + (size_t)NROWS * FEAT;            // 4096*4096
  float* alpha    = P + (size_t)NTOT * NTOT;              // 4096
  float* beta     = alpha + NTOT;                         // 4096
  float* colsum   = beta + NTOT;                          // 4096
  float* bpart    = colsum + NTOT;                        // 64*4096
  float* blockmax = bpart + (size_t)64 * NTOT;            // 16
  float* lossp    = blockmax + 16;                        // 4096
  int*   flags    = (int*)(lossp + NTOT);                 // 3

  const int  wavesPerBlk = 8;
  const dim3 blk(32 * wavesPerBlk);

  // Embeds: 2x2-blocked waves over (4096/32) x (128/32) = 512 wave-blocks.
  const int embedWaves = (NROWS / 32) * (FEAT / 32);      // 512
  embed_gemm_kernel<KT><<<embedWaves / wavesPerBlk, blk, 0, stream>>>(f_t, Wt, bt, et);
  embed_gemm_kernel<KS><<<embedWaves / wavesPerBlk, blk, 0, stream>>>(f_s, Ws, bs, es);
  rownorm_kernel<<<NROWS, 128, 0, stream>>>(et);
  rownorm_kernel<<<NROWS, 128, 0, stream>>>(es);

  // Similarity: 2x4-blocked waves over (4096/32) x (4096/64) = 8192 wave-blocks.
  const int simWaves = (NTOT / 32) * (NTOT / 64);         // 8192
  sim_gemm_kernel<<<simWaves / wavesPerBlk, blk, 0, stream>>>(et, es, Mout);

  p0_kernel<<<NTOT, 256, 0, stream>>>(Mout, P, alpha);
  init_flags_kernel<<<1, 32, 0, stream>>>(flags);

  const float r = 1.0f / (float)NTOT;
  const dim3 csGrid(NTOT / 256, 64);
  for (int it = 0; it < MAXIT_; ++it) {
    colsum_kernel<<<csGrid, 256, 0, stream>>>(P, alpha, bpart, 1);
    beta_reduce_kernel<<<NTOT / 256, 256, 0, stream>>>(bpart, beta, blockmax, r);
    flag_kernel<<<1, 32, 0, stream>>>(blockmax, flags);
    update_kernel<<<NTOT, 256, 0, stream>>>(P, alpha, beta, flags);
  }

  colsum_kernel<<<csGrid, 256, 0, stream>>>(P, alpha, bpart, 0);
  beta_reduce_kernel<<<NTOT / 256, 256, 0, stream>>>(bpart, colsum, blockmax, 1.0f);
  finalize_kernel<<<NTOT, 256, 0, stream>>>(P, colsum, Pout, lossp);
  loss_kernel<<<1, 256, 0, stream>>>(lossp, loss);
}